// PatchedKVCache_10333691314387
// MI455X (gfx1250) — compile-verified
//
#include <hip/hip_runtime.h>
#include <hip/hip_bf16.h>

// Problem shape (fixed by the reference).
#define BB 4
#define HH 32
#define SS 8192
#define DD 128
// Total output elements = 4*32*8192*128 = 134,217,728 (512 MB fp32).
// Max byte offset = 2^29  =>  all offsets fit in signed 32-bit ints.

typedef float v4f __attribute__((ext_vector_type(4)));

// Hardware fp8 path is only visible in the device compilation pass.
#if defined(__HIP_DEVICE_COMPILE__) && defined(__has_builtin)
#  if __has_builtin(__builtin_amdgcn_cvt_pk_fp8_f32) && __has_builtin(__builtin_amdgcn_cvt_f32_fp8)
#    define HAVE_HW_FP8 1
#  endif
#endif

// Quantize to fp8 e4m3 and dequantize back to f32 (RNE, saturate).
// Device pass (probe-confirmed in round 2): uses v_cvt_pk_fp8_f32 /
// v_cvt_f32_fp8. The software body below exists so the host pass can parse
// this __device__ function (never executed on host) and as a safety net.
__device__ __forceinline__ float qdq_e4m3(float x) {
#ifdef HAVE_HW_FP8
    // v_cvt_pk_fp8_f32: packs fp8(x) into byte 0 (word_sel=0).
    int packed = __builtin_amdgcn_cvt_pk_fp8_f32(x, x, 0, false);
    // v_cvt_f32_fp8: byte select 0 (immediate).
    return __builtin_amdgcn_cvt_f32_fp8(packed, 0);
#else
    // Software e4m3fn quant-dequant: 3 mantissa bits, bias 7, max 448,
    // min normal 2^-6 (subnormal quantum 2^-9), RNE.
    if (!(x == x)) return x;                     // NaN passthrough
    float ax = fabsf(x);
    if (ax > 448.0f) ax = 448.0f;                // saturating cast
    int e;
    (void)frexpf(ax, &e);                        // ax = m * 2^e, m in [0.5,1)
    float E = (ax == 0.0f) ? -6.0f : fmaxf((float)(e - 1), -6.0f);
    float step = exp2f(E - 3.0f);                // quantization step
    float q = rintf(ax / step) * step;           // RNE
    if (q > 448.0f) q = 448.0f;
    return copysignf(q, x);
#endif
}

// ---------------------------------------------------------------------------
// Kernel 1: streaming out[i] = cache[i] * s_out, b128 non-temporal both ways.
// 8192 blocks * 256 threads (8 wave32 per block), each thread moves exactly
// 16 float4s with 4-deep load pipelining. 32-bit (int) indexing so the
// backend can use saddr+voffset addressing instead of 64-bit VGPR-pair math.
// ---------------------------------------------------------------------------
__global__ void __launch_bounds__(256)
kv_scale_copy_kernel(const v4f* __restrict__ in, v4f* __restrict__ out,
                     const float* __restrict__ s_out_p, int n4) {
    const float s = *s_out_p;                      // uniform -> s_load_b32
    const int stride = (int)(gridDim.x * blockDim.x);
    int i = (int)(blockIdx.x * blockDim.x + threadIdx.x);

    // Main pipelined loop: 4 outstanding b128 NT loads before any store.
    for (; i + 3 * stride < n4; i += 4 * stride) {
        v4f a = __builtin_nontemporal_load(in + i);
        v4f b = __builtin_nontemporal_load(in + i + stride);
        v4f c = __builtin_nontemporal_load(in + i + 2 * stride);
        v4f d = __builtin_nontemporal_load(in + i + 3 * stride);
        a *= s; b *= s; c *= s; d *= s;
        __builtin_nontemporal_store(a, out + i);
        __builtin_nontemporal_store(b, out + i + stride);
        __builtin_nontemporal_store(c, out + i + 2 * stride);
        __builtin_nontemporal_store(d, out + i + 3 * stride);
    }
    // Tail (unused for the exact 2^25 float4 / 2^21 thread configuration).
    for (; i < n4; i += stride) {
        v4f a = __builtin_nontemporal_load(in + i);
        a *= s;
        __builtin_nontemporal_store(a, out + i);
    }
}

// ---------------------------------------------------------------------------
// Kernel 2: patch the scattered row.
// out[b,h,pos,d] = dequant(fp8_e4m3(cur[b,h,0,d] / s_in)) * s_out
// cur is [B,H,1,D] so its flat index == b*H*D + h*D + d == tid.
// 16384 threads total (64 blocks x 256).
// ---------------------------------------------------------------------------
__global__ void __launch_bounds__(256)
kv_patch_row_kernel(const float* __restrict__ cur,
                    const float* __restrict__ s_in_p,
                    const float* __restrict__ s_out_p,
                    const int* __restrict__ idx_p,
                    float* __restrict__ out) {
    const int tid = (int)(blockIdx.x * blockDim.x + threadIdx.x); // 0..B*H*D-1
    const float s_in = *s_in_p;
    const float s_out = *s_out_p;
    int pos = idx_p[0] - 1;
    pos = pos < 0 ? 0 : (pos > SS - 1 ? SS - 1 : pos);

    const float x = cur[tid] / s_in;
    const float y = qdq_e4m3(x);

    const int bh = tid / DD;          // fused (b*H + h)
    const int d  = tid % DD;
    // bh*S + pos <= 128*8192 = 2^20; *D = 2^27 elements -> fits in int.
    const int off = (bh * SS + pos) * DD + d;
    out[off] = y * s_out;
}

extern "C" void kernel_launch(void* const* d_in, const int* in_sizes, int n_in,
                              void* d_out, int out_size, void* d_ws, size_t ws_size,
                              hipStream_t stream) {
    (void)in_sizes; (void)n_in; (void)out_size; (void)d_ws; (void)ws_size;
    const float* cur    = (const float*)d_in[0];   // [B,H,1,D] f32
    const float* cache  = (const float*)d_in[1];   // [B,H,S,D] f32 (fp8-representable)
    const float* s_in   = (const float*)d_in[2];   // [1] f32
    const float* s_out  = (const float*)d_in[3];   // [1] f32
    const int*   idx    = (const int*)d_in[4];     // [1] i32
    float*       out    = (float*)d_out;           // [B,H,S,D] f32

    const int n4 = (int)(((long long)BB * HH * SS * DD) / 4);  // 33,554,432 float4

    // 8192 blocks x 256 threads = 2,097,152 threads; exactly 16 float4/thread.
    kv_scale_copy_kernel<<<dim3(8192), dim3(256), 0, stream>>>(
        (const v4f*)cache, (v4f*)out, s_out, n4);

    // Overwrite the patched row after the streaming pass (same stream => ordered).
    kv_patch_row_kernel<<<dim3((BB * HH * DD) / 256), dim3(256), 0, stream>>>(
        cur, s_in, s_out, idx, out);
}